// DfOp_36275293782100
// MI455X (gfx1250) — compile-verified
//
#include <hip/hip_runtime.h>
#include <stdint.h>

// Problem constants (from the reference)
#define B_    8
#define T_    3000
#define F_    481
#define NDF   96
#define NORD  5
#define ROWF  (2 * F_)          // 962 floats per (b,t) row of spec/out
#define TT    64                // time-tile per block
#define SROWS (TT + NORD - 1)   // 68 staged time rows (includes 4 history rows)
#define TY    4                 // block = (96, 4) -> 384 threads = 12 waves

// Native 2-float vector so __builtin_nontemporal_* accepts it.
typedef float vfloat2 __attribute__((ext_vector_type(2)));

// -------------------------------------------------------------------------
// Kernel 1: deep-filter the first NDF frequency bins.
// Stages spec[b, t0-4 .. t0+TT-1, 0:NDF, :] into LDS with CDNA5 async
// global->LDS loads (ASYNCcnt), then computes the 5-tap complex FIR.
// Block layout: threadIdx.x = frequency bin f (0..95), threadIdx.y = time
// sub-slot -> no integer div/mod in the hot loops.
// -------------------------------------------------------------------------
__global__ void __launch_bounds__(NDF * TY)
df_kernel(const float* __restrict__ spec,
          const float* __restrict__ coef,
          float* __restrict__ out) {
    __shared__ vfloat2 s[SROWS * NDF];   // 68*96*8 = 52,224 bytes

    const int b  = blockIdx.y;
    const int t0 = blockIdx.x * TT;
    const int f  = threadIdx.x;          // 0..95
    const int ty = threadIdx.y;          // 0..3

    const float* specB = spec + (size_t)b * T_ * ROWF;

    // ---- Stage: async copy the DF region (first 192 floats of each row) ----
    // Each (f, ty) thread covers rows ty, ty+4, ty+8, ... (17 rows each).
    for (int row = ty; row < SROWS; row += TY) {
        const int t = t0 - (NORD - 1) + row;
        const int li = row * NDF + f;
        const uint32_t lds_off = (uint32_t)(uintptr_t)(&s[li]);
        if (t >= 0 && t < T_) {
            const float* g = specB + (size_t)t * ROWF + 2 * f;  // 8B aligned
            // CDNA5 async global->LDS copy, tracked by ASYNCcnt.
            asm volatile("global_load_async_to_lds_b64 %0, %1, off"
                         :: "v"(lds_off),
                            "v"((unsigned long long)(uintptr_t)g)
                         : "memory");
        } else if (t < 0) {
            // zero-padded history before t=0 (rows past T are never read)
            s[li] = (vfloat2){0.0f, 0.0f};
        }
    }

#if __has_builtin(__builtin_amdgcn_s_wait_asynccnt)
    __builtin_amdgcn_s_wait_asynccnt(0);
#else
    asm volatile("s_wait_asynccnt 0x0" ::: "memory");
#endif
    __syncthreads();

    // ---- Compute: 16 (t, f) complex outputs per thread, no div/mod ----
    for (int tl = ty; tl < TT; tl += TY) {
        const int t = t0 + tl;
        if (t >= T_) continue;

        // coef row: 10 contiguous floats, 40B -> 8B aligned; read-once => NT
        const float* cp = coef + ((size_t)(b * T_ + t) * NDF + f) * (2 * NORD);
        const vfloat2 c01 = __builtin_nontemporal_load((const vfloat2*)(cp + 0)); // cr0 cr1
        const vfloat2 c23 = __builtin_nontemporal_load((const vfloat2*)(cp + 2)); // cr2 cr3
        const vfloat2 c45 = __builtin_nontemporal_load((const vfloat2*)(cp + 4)); // cr4 ci0
        const vfloat2 c67 = __builtin_nontemporal_load((const vfloat2*)(cp + 6)); // ci1 ci2
        const vfloat2 c89 = __builtin_nontemporal_load((const vfloat2*)(cp + 8)); // ci3 ci4
        const float cr[NORD] = {c01.x, c01.y, c23.x, c23.y, c45.x};
        const float ci[NORD] = {c45.y, c67.x, c67.y, c89.x, c89.y};

        float fr = 0.0f, fi = 0.0f;
#pragma unroll
        for (int k = 0; k < NORD; ++k) {
            const vfloat2 u = s[(tl + k) * NDF + f];
            fr = fmaf(u.x, cr[k], fr);
            fr = fmaf(-u.y, ci[k], fr);
            fi = fmaf(u.x, ci[k], fi);
            fi = fmaf(u.y, cr[k], fi);
        }

        // write-once stream => NT store
        vfloat2* o = (vfloat2*)(out + (size_t)(b * T_ + t) * ROWF + 2 * f);
        __builtin_nontemporal_store((vfloat2){fr, fi}, o);
    }
}

// -------------------------------------------------------------------------
// Kernel 2: pass-through copy of bins [NDF, F) — contiguous 385 float2 per
// (b,t) row. Pure one-shot stream => NT loads and stores.
// -------------------------------------------------------------------------
__global__ void __launch_bounds__(256)
copy_kernel(const float* __restrict__ spec, float* __restrict__ out) {
    const int NC = F_ - NDF;  // 385 complex values per row
    const size_t total = (size_t)B_ * T_ * NC;
    size_t j = (size_t)blockIdx.x * blockDim.x + threadIdx.x;
    if (j >= total) return;
    const size_t bt = j / NC;
    const int    r  = (int)(j - bt * NC);
    const size_t off = bt * ROWF + 2 * NDF + 2 * (size_t)r;  // 8B aligned
    const vfloat2 v = __builtin_nontemporal_load((const vfloat2*)(spec + off));
    __builtin_nontemporal_store(v, (vfloat2*)(out + off));
}

// -------------------------------------------------------------------------
extern "C" void kernel_launch(void* const* d_in, const int* in_sizes, int n_in,
                              void* d_out, int out_size, void* d_ws, size_t ws_size,
                              hipStream_t stream) {
    (void)in_sizes; (void)n_in; (void)out_size; (void)d_ws; (void)ws_size;
    const float* spec = (const float*)d_in[0];
    const float* coef = (const float*)d_in[1];
    float* out = (float*)d_out;

    dim3 grid1((T_ + TT - 1) / TT, B_);
    df_kernel<<<grid1, dim3(NDF, TY), 0, stream>>>(spec, coef, out);

    const size_t total = (size_t)B_ * T_ * (F_ - NDF);
    const int blocks = (int)((total + 255) / 256);
    copy_kernel<<<blocks, dim3(256), 0, stream>>>(spec, out);
}